// BDC_CLIP_12945031430167
// MI455X (gfx1250) — compile-verified
//
#include <hip/hip_runtime.h>
#include <hip/hip_bf16.h>
#include <math.h>

typedef __attribute__((ext_vector_type(2))) float v2f;
typedef __attribute__((ext_vector_type(8))) float v8f;

#define BT    512
#define NTOK  197
#define NPAD  200         // zero-padded token rows for guard-free WMMA K-loops
#define DDIM  768
#define RDIM  256
#define NF    16
#define NB    32          // BT / NF
#define FDIM  32896       // RDIM*(RDIM+1)/2
#define INV2M (1.0f / (2.0f * 197.0f))

__device__ __forceinline__ v8f wmma_f32(v2f a, v2f b, v8f c) {
    // V_WMMA_F32_16X16X4_F32 : D(16x16 f32) = A(16x4 f32) * B(4x16 f32) + C
    return __builtin_amdgcn_wmma_f32_16x16x4_f32(
        /*neg_a=*/false, a, /*neg_b=*/false, b,
        /*c_mod=*/(short)0, c, /*reuse_a=*/false, /*reuse_b=*/false);
}

// ---------------------------------------------------------------------------
// Kernel 1: one workgroup per frame (bt). LN + projection + softmax weighting
// + BDC pooling + triuvec, entirely LDS-resident except the triuvec output.
// ---------------------------------------------------------------------------
__global__ __launch_bounds__(256)
void bdc_frame_kernel(const float* __restrict__ x,
                      const float* __restrict__ gamma,
                      const float* __restrict__ beta,
                      const float* __restrict__ Wred,
                      float* __restrict__ g_bdc,
                      float* __restrict__ g_cls) {
    extern __shared__ float smem[];
    float* s_red   = smem;                  // NPAD*RDIM (200*256), rows 197..199 zero
    float* s_gamma = s_red + NPAD * RDIM;   // 768
    float* s_beta  = s_gamma + DDIM;        // 768
    float* s_mu    = s_beta + DDIM;         // 224
    float* s_rs    = s_mu + 224;            // 224
    float* s_sc    = s_rs + 224;            // 224
    float* s_w     = s_sc + 224;            // 224
    float* s_diag  = s_w + 224;             // 256
    float* s_rm    = s_diag + RDIM;         // 256
    float* s_scr   = s_rm + RDIM;           // 64

    const int tid  = threadIdx.x;
    const int lane = tid & 31;
    const int wid  = tid >> 5;
    const int lh   = lane >> 4;   // 0/1: which K-half this lane feeds
    const int ll   = lane & 15;
    const int bt   = blockIdx.x;
    const float* xb = x + (size_t)bt * NTOK * DDIM;

    // ---- stage gamma/beta + zero the pad rows of s_red ----
    for (int d = tid; d < DDIM; d += 256) { s_gamma[d] = gamma[d]; s_beta[d] = beta[d]; }
    for (int i = tid; i < (NPAD - NTOK) * RDIM; i += 256) s_red[NTOK * RDIM + i] = 0.f;

    // ---- LayerNorm statistics, one wave per row ----
    for (int m = wid; m < NTOK; m += 8) {
        const float* row = xb + (size_t)m * DDIM;
        float s = 0.f, s2 = 0.f;
        for (int d = lane; d < DDIM; d += 32) { float v = row[d]; s += v; s2 += v * v; }
        for (int o = 16; o > 0; o >>= 1) { s += __shfl_xor(s, o); s2 += __shfl_xor(s2, o); }
        if (lane == 0) {
            float mu  = s / (float)DDIM;
            float var = s2 / (float)DDIM - mu * mu;
            s_mu[m] = mu;
            s_rs[m] = rsqrtf(var + 1e-5f);
        }
    }
    __syncthreads();

    // ---- projection: red = LN(x) @ Wred^T, 13x16 tiles of 16x16, f32 WMMA ----
    // Out-of-range token rows are computed from a clamped source row (their
    // WMMA output rows are discarded at the store), keeping the loop branch-free.
    for (int t = wid; t < 13 * 16; t += 8) {
        int ti = t >> 4, tj = t & 15;
        int m0 = ti * 16, r0 = tj * 16;
        int am = m0 + ll;
        int ams = (am < NTOK) ? am : 0;     // clamp: garbage rows dropped at store
        float mu = s_mu[ams], rs = s_rs[ams];
        const float* xrow = xb + (size_t)ams * DDIM;
        const float* wrow = Wred + (size_t)(r0 + ll) * DDIM;
        __builtin_prefetch(xrow, 0, 1);
        __builtin_prefetch(wrow, 0, 1);
        v8f acc = {};
        for (int k0 = 0; k0 < DDIM; k0 += 4) {
            int ka = k0 + 2 * lh;           // even -> 8B-aligned v2f loads
            v2f xv = *(const v2f*)(xrow + ka);
            v2f gv = *(const v2f*)(s_gamma + ka);
            v2f bv = *(const v2f*)(s_beta + ka);
            v2f b  = *(const v2f*)(wrow + ka);
            v2f a;
            a.x = (xv.x - mu) * rs * gv.x + bv.x;
            a.y = (xv.y - mu) * rs * gv.y + bv.y;
            acc = wmma_f32(a, b, acc);
        }
        for (int v = 0; v < 8; v++) {
            int m = m0 + v + 8 * lh;        // C layout: M = v + 8*(lane>>4), N = lane&15
            if (m < NTOK) s_red[m * RDIM + r0 + ll] = acc[v];
        }
    }
    __syncthreads();

    // ---- cls export (pre-weighting) + attention scores ----
    if (tid < RDIM) g_cls[(size_t)bt * RDIM + tid] = s_red[tid];
    if (tid < NTOK) {
        const float* rn = s_red + tid * RDIM;
        float s = 0.f;
        for (int r = 0; r < RDIM; r++) s += s_red[r] * rn[r];
        s_sc[tid] = s * 0.0625f;                // R^-0.5 = 1/16
    }
    __syncthreads();

    // ---- softmax over 197 tokens ----
    {
        float val = (tid < NTOK) ? s_sc[tid] : -3.4e38f;
        float m = val;
        for (int o = 16; o > 0; o >>= 1) m = fmaxf(m, __shfl_xor(m, o));
        if (lane == 0) s_scr[wid] = m;
        __syncthreads();
        float mx = s_scr[0];
        for (int i = 1; i < 8; i++) mx = fmaxf(mx, s_scr[i]);
        float e = (tid < NTOK) ? __expf(val - mx) : 0.f;
        float ssum = e;
        for (int o = 16; o > 0; o >>= 1) ssum += __shfl_xor(ssum, o);
        if (lane == 0) s_scr[8 + wid] = ssum;
        __syncthreads();
        float tot = 0.f;
        for (int i = 0; i < 8; i++) tot += s_scr[8 + i];
        if (tid < NTOK) s_w[tid] = e / tot;
    }
    __syncthreads();

    // ---- apply weights in place: xw[m][r] = w[m]*red[m][r] (pad rows stay 0) ----
    for (int i = tid; i < NTOK * RDIM; i += 256) s_red[i] *= s_w[i >> 8];
    __syncthreads();

    // ---- diag of Gram + init reductions ----
    if (tid < RDIM) {
        float s = 0.f;
        for (int m = 0; m < NTOK; m++) { float v = s_red[m * RDIM + tid]; s += v * v; }
        s_diag[tid] = s;
        s_rm[tid]   = 0.f;
    }
    if (tid == 0) s_scr[0] = 0.f;
    __syncthreads();

    // ---- pass A: all 256 Gram tiles via WMMA, accumulate dcov row sums + grand sum ----
    for (int t = wid; t < 256; t += 8) {
        int ti = t >> 4, tj = t & 15;
        int d0 = ti * 16, e0 = tj * 16;
        v8f acc = {};
        for (int k0 = 0; k0 < NPAD; k0 += 4) {     // guard-free: rows 197..199 are zero
            int ka = k0 + 2 * lh;
            v2f a, b;
            a.x = s_red[ka * RDIM + d0 + ll];
            a.y = s_red[(ka + 1) * RDIM + d0 + ll];
            b.x = s_red[ka * RDIM + e0 + ll];
            b.y = s_red[(ka + 1) * RDIM + e0 + ll];
            acc = wmma_f32(a, b, acc);
        }
        float de = s_diag[e0 + ll];
        float tsum = 0.f;
        for (int v = 0; v < 8; v++) {
            int d = d0 + v + 8 * lh;
            float val = (s_diag[d] + de - 2.f * acc[v]) * INV2M;
            val = sqrtf(fmaxf(val, 0.f) + 1e-5f);
            float rs = val;
            for (int o = 1; o < 16; o <<= 1) rs += __shfl_xor(rs, o);  // sum over 16 cols
            if (ll == 0) { atomicAdd(&s_rm[d], rs); tsum += rs; }
        }
        if (ll == 0) atomicAdd(&s_scr[0], tsum);
    }
    __syncthreads();
    if (tid < RDIM) s_rm[tid] *= (1.0f / 256.0f);
    if (tid == 0)   s_scr[1] = s_scr[0] * (1.0f / 65536.0f);
    __syncthreads();

    // ---- pass B: recompute upper-triangle tiles, center, write triuvec ----
    float gm = s_scr[1];
    float* outb = g_bdc + (size_t)bt * FDIM;
    for (int t = wid; t < 256; t += 8) {
        int ti = t >> 4, tj = t & 15;
        if (ti > tj) continue;
        int d0 = ti * 16, e0 = tj * 16;
        v8f acc = {};
        for (int k0 = 0; k0 < NPAD; k0 += 4) {
            int ka = k0 + 2 * lh;
            v2f a, b;
            a.x = s_red[ka * RDIM + d0 + ll];
            a.y = s_red[(ka + 1) * RDIM + d0 + ll];
            b.x = s_red[ka * RDIM + e0 + ll];
            b.y = s_red[(ka + 1) * RDIM + e0 + ll];
            acc = wmma_f32(a, b, acc);
        }
        float de  = s_diag[e0 + ll];
        float rme = s_rm[e0 + ll];
        int e = e0 + ll;
        for (int v = 0; v < 8; v++) {
            int d = d0 + v + 8 * lh;
            if (d <= e) {
                float val = (s_diag[d] + de - 2.f * acc[v]) * INV2M;
                val = sqrtf(fmaxf(val, 0.f) + 1e-5f);
                val = val - s_rm[d] - rme + gm;
                int idx = d * RDIM - (d * (d - 1)) / 2 + (e - d);   // triu_indices row-major
                outb[idx] = val;
            }
        }
    }
}

// ---------------------------------------------------------------------------
// Kernel 2: per-batch temporal attention -> mixing coefficients.
// mean_t(attn@v + v) = sum_s coeff[s]*v[s], coeff[s] = (1 + sum_t attn[t][s])/16
// ---------------------------------------------------------------------------
__global__ __launch_bounds__(256)
void temporal_attn_kernel(const float* __restrict__ g_cls,
                          const float* __restrict__ Wq,
                          const float* __restrict__ Wk,
                          float* __restrict__ g_coeff) {
    __shared__ float s_qk[2 * NF * RDIM];   // q then k, 16x256 each
    const int tid  = threadIdx.x;
    const int lane = tid & 31, wid = tid >> 5;
    const int lh   = lane >> 4, ll = lane & 15;
    const int b    = blockIdx.x;
    const float* clsb = g_cls + (size_t)b * NF * RDIM;

    // 32 WMMA tile jobs: sel (q/k) x 16 column tiles
    for (int j = wid; j < 32; j += 8) {
        int sel = j >> 4, tj = j & 15;
        const float* Wm = sel ? Wk : Wq;
        int r0 = tj * 16;
        const float* arow = clsb + (size_t)ll * RDIM;
        const float* wrow = Wm + (size_t)(r0 + ll) * RDIM;
        v8f acc = {};
        for (int k0 = 0; k0 < RDIM; k0 += 4) {
            int ka = k0 + 2 * lh;
            v2f a  = *(const v2f*)(arow + ka);
            v2f bb = *(const v2f*)(wrow + ka);
            acc = wmma_f32(a, bb, acc);
        }
        float* dst = s_qk + sel * NF * RDIM;
        for (int v = 0; v < 8; v++) dst[(v + 8 * lh) * RDIM + r0 + ll] = acc[v];
    }
    __syncthreads();

    if (wid == 0) {
        const float* sq = s_qk;
        const float* sk = s_qk + NF * RDIM;
        v8f acc = {};
        for (int k0 = 0; k0 < RDIM; k0 += 4) {
            int ka = k0 + 2 * lh;
            v2f a  = *(const v2f*)(sq + ll * RDIM + ka);
            v2f bb = *(const v2f*)(sk + ll * RDIM + ka);
            acc = wmma_f32(a, bb, acc);
        }
        // row-wise softmax (t = v+8*lh over s = ll), then column sums over t
        float csum_half = 0.f;
        for (int v = 0; v < 8; v++) {
            float sv = acc[v] * 0.0625f;
            float m = sv;
            for (int o = 1; o < 16; o <<= 1) m = fmaxf(m, __shfl_xor(m, o));
            float e = __expf(sv - m);
            float ssum = e;
            for (int o = 1; o < 16; o <<= 1) ssum += __shfl_xor(ssum, o);
            csum_half += e / ssum;
        }
        float col = csum_half + __shfl_xor(csum_half, 16);
        if (lh == 0) g_coeff[(size_t)b * NF + ll] = (col + 1.0f) * (1.0f / 16.0f);
    }
}

// ---------------------------------------------------------------------------
// Kernel 3: out[b][f] = sum_s coeff[b][s] * frame_bdc[b*16+s][f]  (coalesced)
// ---------------------------------------------------------------------------
__global__ __launch_bounds__(256)
void temporal_mix_kernel(const float* __restrict__ g_bdc,
                         const float* __restrict__ g_coeff,
                         float* __restrict__ out) {
    int i = blockIdx.x * 256 + threadIdx.x;
    if (i >= NB * FDIM) return;
    int b = i / FDIM;
    int f = i - b * FDIM;
    const float* cf = g_coeff + b * NF;
    const float* vb = g_bdc + (size_t)b * NF * FDIM + f;
    float acc = 0.f;
    for (int s = 0; s < NF; s++) acc += cf[s] * vb[(size_t)s * FDIM];
    out[i] = acc;
}

extern "C" void kernel_launch(void* const* d_in, const int* in_sizes, int n_in,
                              void* d_out, int out_size, void* d_ws, size_t ws_size,
                              hipStream_t stream) {
    const float* x     = (const float*)d_in[0];  // (512,197,768)
    const float* gamma = (const float*)d_in[1];  // (768,)
    const float* beta  = (const float*)d_in[2];  // (768,)
    const float* Wred  = (const float*)d_in[3];  // (256,768)
    const float* Wq    = (const float*)d_in[4];  // (256,256)
    const float* Wk    = (const float*)d_in[5];  // (256,256)
    (void)in_sizes; (void)n_in; (void)out_size; (void)ws_size; // num_frames==16 (d_in[6])

    float* g_bdc   = (float*)d_ws;                      // 512*32896 floats
    float* g_cls   = g_bdc + (size_t)BT * FDIM;         // 512*256 floats
    float* g_coeff = g_cls + (size_t)BT * RDIM;         // 32*16 floats

    size_t shmem1 = (size_t)(NPAD * RDIM + 2 * DDIM + 4 * 224 + 2 * RDIM + 64) * sizeof(float);

    bdc_frame_kernel<<<BT, 256, shmem1, stream>>>(x, gamma, beta, Wred, g_bdc, g_cls);
    temporal_attn_kernel<<<NB, 256, 0, stream>>>(g_cls, Wq, Wk, g_coeff);
    int total = NB * FDIM;
    temporal_mix_kernel<<<(total + 255) / 256, 256, 0, stream>>>(g_bdc, g_coeff, (float*)d_out);
}